// LMulLanguageModel_62234076119647
// MI455X (gfx1250) — compile-verified
//
#include <hip/hip_runtime.h>
#include <math.h>

// ---------------------------------------------------------------------------
// Model constants (match reference setup_inputs)
// ---------------------------------------------------------------------------
#define KD   128   // d_model
#define KS   128   // seq len
#define KB   2     // batch
#define KH   4     // heads
#define KDK  32    // head dim
#define KF   512   // ffn dim
#define KV   1024  // vocab
#define KL   2     // layers
#define LN_EPS 1e-5f
#define MB_BITS 3

typedef float v2f __attribute__((ext_vector_type(2)));
typedef float v8f __attribute__((ext_vector_type(8)));

// ---------------------------------------------------------------------------
// x = emb[tokens] + pos
// ---------------------------------------------------------------------------
__global__ void embed_kernel(const int* __restrict__ tok,
                             const float* __restrict__ emb,
                             const float* __restrict__ pos,
                             float* __restrict__ x, int total) {
  for (int i = blockIdx.x * blockDim.x + threadIdx.x; i < total;
       i += gridDim.x * blockDim.x) {
    int d  = i % KD;
    int bs = i / KD;
    int s  = bs % KS;
    x[i] = emb[tok[bs] * KD + d] + pos[s * KD + d];
  }
}

// ---------------------------------------------------------------------------
// l_mul operand transform: y = round(frexp_m(x)*2^mb) * 2^(2e - mb)
// (so that l_mul(x, wT) == transform(x) @ transform(wT) exactly)
// ---------------------------------------------------------------------------
__global__ void transform_kernel(const float* __restrict__ in,
                                 float* __restrict__ out, int n) {
  for (int i = blockIdx.x * blockDim.x + threadIdx.x; i < n;
       i += gridDim.x * blockDim.x) {
    float xv = in[i];
    int e;
    float m = frexpf(xv, &e);             // x = m * 2^e, 0.5 <= |m| < 1
    float r = roundf(m * 8.0f);           // quantized mantissa, mb=3
    out[i] = ldexpf(r, 2 * e - MB_BITS);  // r * 2^(2e - mb)
  }
}

// ---------------------------------------------------------------------------
// Generic batched f32 WMMA GEMM:  C = scale * (A @ opB(B)) (+bias) (+ReLU)
//   A: [M x K] row-major (lda), per-z offset bz*sAb + hz*sAh  (z = bz*Hdim+hz)
//   B: if TRANSB, B(k,n) = B[n*ldb + k]  else  B[k*ldb + n]
//   C: [M x N] row-major (ldc)
//   NFULL: compile-time guarantee that N % 64 == 0 (no column guards at all)
// Block = 128 threads (4 waves). Each wave computes one 16x16 C tile along N.
// A (16xKC) and B (KCx64) panels staged through LDS; inner loop is
// branch-free: ds_load_b64 x2 + v_wmma_f32_16x16x4_f32.
// Fragment layouts per CDNA5 ISA 7.12.2:
//   A: lane -> M = lane%16 ; vgpr j, half h=lane/16 -> K = j + 2h
//   B: lane -> N = lane%16 ; vgpr j, half h         -> K = j + 2h
//   C: lane -> N = lane%16 ; vgpr j, half h         -> M = j + 8h
// ---------------------------------------------------------------------------
template <int TRANSB, int HAS_BIAS, int RELU, int NFULL>
__global__ __launch_bounds__(128)
void wmma_gemm_kernel(const float* __restrict__ A, int lda, long long sAb, long long sAh,
                      const float* __restrict__ B, int ldb, long long sBb, long long sBh,
                      float* __restrict__ C, int ldc, long long sCb, long long sCh,
                      const float* __restrict__ bias, float scale,
                      int M, int N, int K, int Hdim) {
  constexpr int KC = 32;
  __shared__ float As[16 * KC];        // A panel, row-major [16][KC]
  __shared__ float Bs[(KC / 2) * 128]; // B panel, k-paired: Bs[k/2][n][k&1]

  const int tid  = threadIdx.x;
  const int lane = tid & 31;
  const int wave = tid >> 5;           // 0..3
  const int half = lane >> 4;          // 0/1
  const int lm   = lane & 15;

  const int m0    = blockIdx.x * 16;
  const int n0blk = blockIdx.y * 64;
  const int n0    = n0blk + wave * 16;

  const int z  = blockIdx.z;
  const int bz = z / Hdim;
  const int hz = z % Hdim;
  const float* Ab = A + bz * sAb + hz * sAh;
  const float* Bb = B + bz * sBb + hz * sBh;
  float*       Cb = C + bz * sCb + hz * sCh;

  v8f acc = {};

  for (int kb = 0; kb < K; kb += KC) {
    // --- cooperative A panel load: 16 x 32 floats, one float4 per thread ---
    {
      const int r = tid >> 3;          // 0..15
      const int c = (tid & 7) << 2;    // 0,4,...,28
      const float4* src =
          reinterpret_cast<const float4*>(Ab + (size_t)(m0 + r) * lda + kb + c);
      *reinterpret_cast<float4*>(&As[r * KC + c]) = *src;
      if (kb + KC < K)  // prefetch next A chunk (global_prefetch_b8)
        __builtin_prefetch(Ab + (size_t)(m0 + r) * lda + kb + KC + c, 0, 0);
    }
    // --- cooperative B panel load: 32 x 64 floats (zero-pad n >= N) ---
#pragma unroll
    for (int i = 0; i < 4; ++i) {
      const int idx = i * 128 + tid;   // 0..511 float4 slots
      if (TRANSB) {
        const int nl = idx & 63;           // local n
        const int kg = (idx >> 6) << 2;    // 0,4,...,28
        const int n  = n0blk + nl;
        float4 val = make_float4(0.f, 0.f, 0.f, 0.f);
        if (NFULL || n < N)
          val = *reinterpret_cast<const float4*>(Bb + (size_t)n * ldb + kb + kg);
        *reinterpret_cast<float2*>(&Bs[(kg >> 1) * 128 + nl * 2]) =
            make_float2(val.x, val.y);
        *reinterpret_cast<float2*>(&Bs[((kg >> 1) + 1) * 128 + nl * 2]) =
            make_float2(val.z, val.w);
      } else {
        const int kk = idx >> 4;           // 0..31
        const int n4 = (idx & 15) << 2;    // 0,4,...,60
        const int n  = n0blk + n4;
        float4 val = make_float4(0.f, 0.f, 0.f, 0.f);
        if (NFULL || n < N)
          val = *reinterpret_cast<const float4*>(Bb + (size_t)(kb + kk) * ldb + n);
        const int p = (kk >> 1) * 128 + (kk & 1);
        Bs[p + (n4 + 0) * 2] = val.x;
        Bs[p + (n4 + 1) * 2] = val.y;
        Bs[p + (n4 + 2) * 2] = val.z;
        Bs[p + (n4 + 3) * 2] = val.w;
      }
    }
    __syncthreads();

    // --- 8 branch-free WMMA steps over this K chunk ---
#pragma unroll
    for (int kk = 0; kk < KC; kk += 4) {
      const v2f af =
          *reinterpret_cast<const v2f*>(&As[lm * KC + kk + 2 * half]);
      const v2f bf = *reinterpret_cast<const v2f*>(
          &Bs[((kk + 2 * half) >> 1) * 128 + (wave * 16 + lm) * 2]);
      acc = __builtin_amdgcn_wmma_f32_16x16x4_f32(
          /*neg_a=*/false, af, /*neg_b=*/false, bf,
          /*c_mod=*/(short)0, acc, /*reuse_a=*/false, /*reuse_b=*/false);
    }
    __syncthreads();
  }

  if (NFULL || n0 < N) {
#pragma unroll
    for (int j = 0; j < 8; ++j) {
      const int row = m0 + j + 8 * half;
      const int col = n0 + lm;
      float vv = acc[j] * scale;
      if (HAS_BIAS) vv += bias[col];
      if (RELU) vv = fmaxf(vv, 0.0f);
      Cb[(size_t)row * ldc + col] = vv;
    }
  }
}

// ---------------------------------------------------------------------------
// Row softmax over the last axis (row length = KS = 128 = blockDim.x)
// ---------------------------------------------------------------------------
__global__ __launch_bounds__(128)
void softmax_kernel(float* __restrict__ sc) {
  __shared__ float red[128];
  const int row = blockIdx.x;
  const int tid = threadIdx.x;
  const size_t base = (size_t)row * KS;
  float v = sc[base + tid];

  red[tid] = v;
  __syncthreads();
  for (int s = 64; s > 0; s >>= 1) {
    if (tid < s) red[tid] = fmaxf(red[tid], red[tid + s]);
    __syncthreads();
  }
  const float mx = red[0];
  __syncthreads();

  const float e = expf(v - mx);
  red[tid] = e;
  __syncthreads();
  for (int s = 64; s > 0; s >>= 1) {
    if (tid < s) red[tid] += red[tid + s];
    __syncthreads();
  }
  sc[base + tid] = e / red[0];
}

// ---------------------------------------------------------------------------
// x = LayerNorm(x + a) * g + b     (row width = KD = 128 = blockDim.x)
// ---------------------------------------------------------------------------
__global__ __launch_bounds__(128)
void residual_ln_kernel(float* __restrict__ x, const float* __restrict__ a,
                        const float* __restrict__ g, const float* __restrict__ b) {
  __shared__ float red[128];
  const int row = blockIdx.x;
  const int tid = threadIdx.x;
  const size_t idx = (size_t)row * KD + tid;
  const float v = x[idx] + a[idx];

  red[tid] = v;
  __syncthreads();
  for (int s = 64; s > 0; s >>= 1) {
    if (tid < s) red[tid] += red[tid + s];
    __syncthreads();
  }
  const float mu = red[0] * (1.0f / KD);
  __syncthreads();

  const float d = v - mu;
  red[tid] = d * d;
  __syncthreads();
  for (int s = 64; s > 0; s >>= 1) {
    if (tid < s) red[tid] += red[tid + s];
    __syncthreads();
  }
  const float var = red[0] * (1.0f / KD);
  x[idx] = d * rsqrtf(var + LN_EPS) * g[tid] + b[tid];
}

// ---------------------------------------------------------------------------
// Host launchers
// ---------------------------------------------------------------------------
static inline void launch_transform(const float* in, float* out, int n,
                                    hipStream_t s) {
  int blocks = (n + 255) / 256;
  if (blocks > 512) blocks = 512;
  transform_kernel<<<blocks, 256, 0, s>>>(in, out, n);
}

// Dispatch only the four (TRANSB, HAS_BIAS, RELU, NFULL) combos this model
// uses, to keep the device binary lean:
//   trans GEMMs always have N % 64 == 0  -> NFULL=1
//   the only no-trans GEMM (P @ V, N=32) -> NFULL=0, no bias/relu
static inline void launch_gemm(const float* A, int lda, long long sAb, long long sAh,
                               const float* B, int ldb, long long sBb, long long sBh,
                               int transB,
                               float* C, int ldc, long long sCb, long long sCh,
                               const float* bias, int relu, float scale,
                               int M, int N, int K, int Z, int Hdim,
                               hipStream_t s) {
  dim3 grid(M / 16, (N + 63) / 64, Z);
  dim3 blk(128, 1, 1);
  if (transB) {
    if (bias) {
      if (relu)
        wmma_gemm_kernel<1, 1, 1, 1><<<grid, blk, 0, s>>>(
            A, lda, sAb, sAh, B, ldb, sBb, sBh, C, ldc, sCb, sCh, bias, scale, M, N, K, Hdim);
      else
        wmma_gemm_kernel<1, 1, 0, 1><<<grid, blk, 0, s>>>(
            A, lda, sAb, sAh, B, ldb, sBb, sBh, C, ldc, sCb, sCh, bias, scale, M, N, K, Hdim);
    } else {
      wmma_gemm_kernel<1, 0, 0, 1><<<grid, blk, 0, s>>>(
          A, lda, sAb, sAh, B, ldb, sBb, sBh, C, ldc, sCb, sCh, bias, scale, M, N, K, Hdim);
    }
  } else {
    wmma_gemm_kernel<0, 0, 0, 0><<<grid, blk, 0, s>>>(
        A, lda, sAb, sAh, B, ldb, sBb, sBh, C, ldc, sCb, sCh, bias, scale, M, N, K, Hdim);
  }
}

extern "C" void kernel_launch(void* const* d_in, const int* in_sizes, int n_in,
                              void* d_out, int out_size, void* d_ws, size_t ws_size,
                              hipStream_t stream) {
  (void)in_sizes; (void)n_in; (void)out_size; (void)ws_size;

  const int*   tok  = (const int*)  d_in[0];
  const float* emb  = (const float*)d_in[1];
  const float* pos  = (const float*)d_in[2];
  const float* Wq   = (const float*)d_in[3];
  const float* bq   = (const float*)d_in[4];
  const float* Wk   = (const float*)d_in[5];
  const float* bk   = (const float*)d_in[6];
  const float* Wv   = (const float*)d_in[7];
  const float* bv   = (const float*)d_in[8];
  const float* Wo   = (const float*)d_in[9];
  const float* bo   = (const float*)d_in[10];
  const float* W1   = (const float*)d_in[11];
  const float* b1   = (const float*)d_in[12];
  const float* W2   = (const float*)d_in[13];
  const float* b2   = (const float*)d_in[14];
  const float* g1   = (const float*)d_in[15];
  const float* be1  = (const float*)d_in[16];
  const float* g2   = (const float*)d_in[17];
  const float* be2  = (const float*)d_in[18];
  const float* Wout = (const float*)d_in[19];
  const float* bout = (const float*)d_in[20];
  float* out = (float*)d_out;

  const int M   = KB * KS;   // 256 token rows
  const int DD  = KD * KD;   // 16384
  const int FD  = KF * KD;   // 65536

  // ---- workspace carve (floats) ----
  float* ws = (float*)d_ws;
  float* x    = ws; ws += M * KD;            // hidden state
  float* act  = ws; ws += M * KD;            // attn / ffn output (pre-residual)
  float* h1   = ws; ws += M * KF;            // ffn hidden (post-ReLU)
  float* xT   = ws; ws += M * KF;            // transformed activations (max 256x512)
  float* q    = ws; ws += M * KD;
  float* k    = ws; ws += M * KD;
  float* v    = ws; ws += M * KD;
  float* P    = ws; ws += KB * KH * KS * KS; // attention scores/probs
  float* Wt   = ws;                          // transformed weights
  const int perLayer = 4 * DD + 2 * FD;      // Wq,Wk,Wv,Wo,W1,W2
  float* WoutT = Wt + KL * perLayer;

  // ---- transform all weights (deterministic each call) ----
  for (int l = 0; l < KL; ++l) {
    float* base = Wt + l * perLayer;
    launch_transform(Wq + l * DD, base + 0 * DD, DD, stream);
    launch_transform(Wk + l * DD, base + 1 * DD, DD, stream);
    launch_transform(Wv + l * DD, base + 2 * DD, DD, stream);
    launch_transform(Wo + l * DD, base + 3 * DD, DD, stream);
    launch_transform(W1 + l * FD, base + 4 * DD, FD, stream);
    launch_transform(W2 + l * FD, base + 4 * DD + FD, FD, stream);
  }
  launch_transform(Wout, WoutT, KV * KD, stream);

  // ---- embedding + positional encoding ----
  embed_kernel<<<(M * KD + 255) / 256, 256, 0, stream>>>(tok, emb, pos, x, M * KD);

  const float inv_sqrt_dk = 0.17677669529663687f;  // 1/sqrt(32)
  const long long SD = (long long)KS * KD;
  const long long SS = (long long)KS * KS;

  for (int l = 0; l < KL; ++l) {
    float* WqT = Wt + l * perLayer + 0 * DD;
    float* WkT = Wt + l * perLayer + 1 * DD;
    float* WvT = Wt + l * perLayer + 2 * DD;
    float* WoT = Wt + l * perLayer + 3 * DD;
    float* W1T = Wt + l * perLayer + 4 * DD;
    float* W2T = Wt + l * perLayer + 4 * DD + FD;

    // --- q,k,v = l_mul(x, W*.T) + b* ---
    launch_transform(x, xT, M * KD, stream);
    launch_gemm(xT, KD, 0, 0, WqT, KD, 0, 0, 1, q, KD, 0, 0,
                bq + l * KD, 0, 1.0f, M, KD, KD, 1, 1, stream);
    launch_gemm(xT, KD, 0, 0, WkT, KD, 0, 0, 1, k, KD, 0, 0,
                bk + l * KD, 0, 1.0f, M, KD, KD, 1, 1, stream);
    launch_gemm(xT, KD, 0, 0, WvT, KD, 0, 0, 1, v, KD, 0, 0,
                bv + l * KD, 0, 1.0f, M, KD, KD, 1, 1, stream);

    // --- scores[z] = q[b,h] @ k[b,h]^T / sqrt(dk),  z = b*H + h ---
    launch_gemm(q, KD, SD, KDK, k, KD, SD, KDK, 1, P, KS, KH * SS, SS,
                nullptr, 0, inv_sqrt_dk, KS, KS, KDK, KB * KH, KH, stream);

    softmax_kernel<<<KB * KH * KS, 128, 0, stream>>>(P);

    // --- ctx[b,h] = P[z] @ v[b,h]  (written head-concatenated into q buffer) ---
    launch_gemm(P, KS, KH * SS, SS, v, KD, SD, KDK, 0, q, KD, SD, KDK,
                nullptr, 0, 1.0f, KS, KDK, KS, KB * KH, KH, stream);

    // --- attn out = l_mul(ctx, Wo.T) + bo ---
    launch_transform(q, xT, M * KD, stream);
    launch_gemm(xT, KD, 0, 0, WoT, KD, 0, 0, 1, act, KD, 0, 0,
                bo + l * KD, 0, 1.0f, M, KD, KD, 1, 1, stream);

    // --- x = LN(x + attn) ---
    residual_ln_kernel<<<M, 128, 0, stream>>>(x, act, g1 + l * KD, be1 + l * KD);

    // --- h1 = relu(l_mul(x, W1.T) + b1) ---
    launch_transform(x, xT, M * KD, stream);
    launch_gemm(xT, KD, 0, 0, W1T, KD, 0, 0, 1, h1, KF, 0, 0,
                b1 + l * KF, 1, 1.0f, M, KF, KD, 1, 1, stream);

    // --- ff = l_mul(h1, W2.T) + b2 ---
    launch_transform(h1, xT, M * KF, stream);
    launch_gemm(xT, KF, 0, 0, W2T, KF, 0, 0, 1, act, KD, 0, 0,
                b2 + l * KD, 0, 1.0f, M, KD, KF, 1, 1, stream);

    // --- x = LN(x + ff) ---
    residual_ln_kernel<<<M, 128, 0, stream>>>(x, act, g2 + l * KD, be2 + l * KD);
  }

  // --- logits = l_mul(x, Wout.T) + bout ---
  launch_transform(x, xT, M * KD, stream);
  launch_gemm(xT, KD, 0, 0, WoutT, KD, 0, 0, 1, out, KV, 0, 0,
              bout, 0, 1.0f, M, KV, KD, 1, 1, stream);
}